// GCN_15015205667144
// MI455X (gfx1250) — compile-verified
//
#include <hip/hip_runtime.h>
#include <hip/hip_bf16.h>

// ---------------------------------------------------------------------------
// GCN forward on gfx1250 (MI455X):
//   h0 = x @ W0                         (WMMA f32 16x16x4, B staged in LDS)
//   h1 = relu(spmm(adj, h0) + b0)       (sorted-row segments, async-LDS staging)
//   mean/scale = batchnorm stats(h1)    (two-stage deterministic reduction)
//   g1 = ((h1-mean)*scale) @ W1         (WMMA f32 16x16x4, W1+stats in LDS)
//   out = spmm(adj, g1) + b1
// ---------------------------------------------------------------------------

#define N_NODES 100000
#define N_EDGES 1600000
#define D_IN    512
#define D_H     128
#define D_OUT   64
#define BN_EPS  1e-5f
#define NB_STAT 500   // blocks for BN partial reduction
#define KC      32    // K-chunk staged in LDS for GEMM0
#define MTILES  (N_NODES / 16)   // 6250 16-row WMMA M-tiles

typedef float v2f __attribute__((ext_vector_type(2)));
typedef float v8f __attribute__((ext_vector_type(8)));

// ---------------------------------------------------------------------------
// GEMM0: H0[100000,128] = X[100000,512] @ W0[512,128]
// Block = 8 waves, each wave one 16-row M-tile x all 128 columns
// (8 accumulators = 64 VGPRs -> no spills). W0 is staged chunk-by-chunk
// (KC x 128 = 16KB) into LDS so all 8 waves share one global read of W0;
// the hot loop issues only 2 A-loads (VMEM) + 16 B-loads (DS) per 8 WMMAs.
// A frag (16x4 f32): lanes 0-15 -> K=k0,k0+1 ; lanes 16-31 -> K=k0+2,k0+3
// B frag (4x16 f32): same K split, N = lane&15 within each 16-col tile.
// ---------------------------------------------------------------------------
__global__ __launch_bounds__(256) void gcn_gemm0(const float* __restrict__ X,
                                                 const float* __restrict__ W0,
                                                 float* __restrict__ H0) {
    __shared__ float sB[KC * D_H];          // 16 KB

    const int wtile = threadIdx.x >> 5;     // wave index in block: 0..7
    const int lane  = threadIdx.x & 31;
    const int half  = lane >> 4;            // 0: K k0,k0+1   1: K k0+2,k0+3
    const int lrow  = lane & 15;
    const int nRB   = (MTILES + 7) / 8;     // 782 row-blocks of 8 M-tiles

    for (int rb = blockIdx.x; rb < nRB; rb += gridDim.x) {
        const int  mt    = rb * 8 + wtile;          // this wave's 16-row tile
        const bool valid = mt < MTILES;             // wave-uniform
        const float* __restrict__ xrow =
            X + (size_t)((valid ? mt : 0) * 16 + lrow) * D_IN;

        v8f acc[8];
#pragma unroll
        for (int nt = 0; nt < 8; ++nt) acc[nt] = (v8f)0.f;

        for (int kc = 0; kc < D_IN; kc += KC) {
            // cooperative, coalesced stage of W0[kc..kc+KC) (row-major, linear)
            {
                const float4* __restrict__ src =
                    (const float4*)(W0 + (size_t)kc * D_H);
                float4* dst = (float4*)sB;
#pragma unroll
                for (int i = 0; i < (KC * D_H / 4) / 256; ++i)
                    dst[threadIdx.x + 256 * i] = src[threadIdx.x + 256 * i];
            }
            __syncthreads();

            if (valid) {
#pragma unroll
                for (int kk = 0; kk < KC / 4; ++kk) {
                    const int k0 = kk * 4 + half * 2;
                    v2f a;
                    a.x = xrow[kc + k0];
                    a.y = xrow[kc + k0 + 1];
                    const float* __restrict__ bk = sB + k0 * D_H;
#pragma unroll
                    for (int nt = 0; nt < 8; ++nt) {
                        const int cc = nt * 16 + lrow;
                        v2f b;
                        b.x = bk[cc];
                        b.y = bk[D_H + cc];
                        acc[nt] = __builtin_amdgcn_wmma_f32_16x16x4_f32(
                            false, a, false, b, (short)0, acc[nt], false, false);
                    }
                }
            }
            __syncthreads();
        }
        if (valid) {
            // C/D layout: VGPR r -> M = half*8 + r ; N = nt*16 + lrow
#pragma unroll
            for (int nt = 0; nt < 8; ++nt) {
#pragma unroll
                for (int r = 0; r < 8; ++r) {
                    const int ro = mt * 16 + half * 8 + r;
                    const int co = nt * 16 + lrow;
                    H0[(size_t)ro * D_H + co] = acc[nt][r];
                }
            }
        }
    }
}

// ---------------------------------------------------------------------------
// GEMM1: G1[100000,64] = ((H1 - mean) * scale) @ W1[128,64]
// Whole W1 (32KB) + BN stats staged in LDS once; normalization fused into
// the A-fragment load. 4 accumulators per wave, no barriers in the K-loop.
// ---------------------------------------------------------------------------
__global__ __launch_bounds__(256) void gcn_gemm1(const float* __restrict__ H1,
                                                 const float* __restrict__ mean,
                                                 const float* __restrict__ scale,
                                                 const float* __restrict__ W1,
                                                 float* __restrict__ G1) {
    __shared__ float sW[D_H * D_OUT];       // 32 KB
    __shared__ float sM[D_H];
    __shared__ float sS[D_H];

    for (int i = threadIdx.x; i < D_H * D_OUT; i += 256) sW[i] = W1[i];
    if (threadIdx.x < D_H) {
        sM[threadIdx.x] = mean[threadIdx.x];
        sS[threadIdx.x] = scale[threadIdx.x];
    }
    __syncthreads();

    const int wave   = (blockIdx.x * blockDim.x + threadIdx.x) >> 5;
    const int nwaves = (gridDim.x * blockDim.x) >> 5;
    const int lane   = threadIdx.x & 31;
    const int half   = lane >> 4;
    const int lrow   = lane & 15;

    for (int mt = wave; mt < MTILES; mt += nwaves) {
        const float* __restrict__ hr = H1 + (size_t)(mt * 16 + lrow) * D_H;
        v8f acc[4];
#pragma unroll
        for (int nt = 0; nt < 4; ++nt) acc[nt] = (v8f)0.f;

        for (int kk = 0; kk < D_H / 4; ++kk) {
            const int k0 = kk * 4 + half * 2;
            v2f a;
            a.x = (hr[k0]     - sM[k0])     * sS[k0];
            a.y = (hr[k0 + 1] - sM[k0 + 1]) * sS[k0 + 1];
            const float* __restrict__ bk = sW + k0 * D_OUT;
#pragma unroll
            for (int nt = 0; nt < 4; ++nt) {
                const int cc = nt * 16 + lrow;
                v2f b;
                b.x = bk[cc];
                b.y = bk[D_OUT + cc];
                acc[nt] = __builtin_amdgcn_wmma_f32_16x16x4_f32(
                    false, a, false, b, (short)0, acc[nt], false, false);
            }
        }
#pragma unroll
        for (int nt = 0; nt < 4; ++nt) {
#pragma unroll
            for (int r = 0; r < 8; ++r) {
                const int ro = mt * 16 + half * 8 + r;
                const int co = nt * 16 + lrow;
                G1[(size_t)ro * D_OUT + co] = acc[nt][r];
            }
        }
    }
}

// ---------------------------------------------------------------------------
// SpMM: out[n,:] = sum_{e: row[e]==n} vals[e] * H[col[e],:]  (+ bias [, relu])
// row[] is sorted -> binary search the segment, deterministic reduction.
// One block per node; each thread owns one feature column. (col,val) chunks
// are staged into LDS with GLOBAL_LOAD_ASYNC_TO_LDS (ASYNCcnt-tracked),
// waited with s_wait_asynccnt before the workgroup barrier. The async op's
// VDST is a single 32-bit VGPR holding the LDS byte offset (low 32 bits of
// the LDS-aperture address).
// ---------------------------------------------------------------------------
__device__ __forceinline__ int lower_bound_i32(const int* __restrict__ a,
                                               int n, int key) {
    int lo = 0, hi = n;
    while (lo < hi) {
        int mid = (lo + hi) >> 1;
        if (a[mid] < key) lo = mid + 1; else hi = mid;
    }
    return lo;
}

template <int D, bool RELU>
__global__ __launch_bounds__(D) void gcn_spmm(const float* __restrict__ vals,
                                              const int* __restrict__ rowi,
                                              const int* __restrict__ coli,
                                              const float* __restrict__ H,
                                              const float* __restrict__ bias,
                                              float* __restrict__ O) {
    const int node = blockIdx.x;
    const int tid  = threadIdx.x;

    __shared__ int   s_lo, s_hi;
    __shared__ int   s_col[64];
    __shared__ float s_val[64];

    if (tid == 0) {
        s_lo = lower_bound_i32(rowi, N_EDGES, node);
        s_hi = lower_bound_i32(rowi, N_EDGES, node + 1);
    }
    __syncthreads();
    const int lo = s_lo, hi = s_hi;

    float acc = 0.f;
    for (int base = lo; base < hi; base += 64) {
        int cnt = hi - base;
        if (cnt > 64) cnt = 64;
        __syncthreads();   // prior-iteration LDS reads complete
        if (tid < cnt) {
            const unsigned lds_col = (unsigned)(unsigned long long)(&s_col[tid]);
            const unsigned lds_val = (unsigned)(unsigned long long)(&s_val[tid]);
            asm volatile("global_load_async_to_lds_b32 %0, %1, off"
                         :: "v"(lds_col), "v"(coli + base + tid)
                         : "memory");
            asm volatile("global_load_async_to_lds_b32 %0, %1, off"
                         :: "v"(lds_val), "v"(vals + base + tid)
                         : "memory");
        }
        asm volatile("s_wait_asynccnt 0x0" ::: "memory");
        __syncthreads();
        for (int i = 0; i < cnt; ++i)
            acc = fmaf(s_val[i], H[(size_t)s_col[i] * D + tid], acc);
    }
    float r = acc + bias[tid];
    if (RELU) r = fmaxf(r, 0.f);
    O[(size_t)node * D + tid] = r;
}

// ---------------------------------------------------------------------------
// BatchNorm batch statistics (biased var), two-stage deterministic reduction.
// ---------------------------------------------------------------------------
__global__ __launch_bounds__(256) void gcn_bn_stats(const float* __restrict__ H1,
                                                    float* __restrict__ psum,
                                                    float* __restrict__ psq) {
    const int c = threadIdx.x & (D_H - 1);
    const int g = threadIdx.x >> 7;                 // 0 or 1
    const int rows_per_block = (N_NODES + gridDim.x - 1) / gridDim.x;
    const int r0 = blockIdx.x * rows_per_block;
    int r1 = r0 + rows_per_block;
    if (r1 > N_NODES) r1 = N_NODES;

    float s = 0.f, q = 0.f;
    for (int r = r0 + g; r < r1; r += 2) {
        const float v = H1[(size_t)r * D_H + c];
        s += v;
        q += v * v;
    }
    __shared__ float ss[256], sq[256];
    ss[threadIdx.x] = s;
    sq[threadIdx.x] = q;
    __syncthreads();
    if (threadIdx.x < D_H) {
        psum[(size_t)blockIdx.x * D_H + c] = ss[c] + ss[c + D_H];
        psq [(size_t)blockIdx.x * D_H + c] = sq[c] + sq[c + D_H];
    }
}

__global__ __launch_bounds__(D_H) void gcn_bn_finalize(const float* __restrict__ psum,
                                                       const float* __restrict__ psq,
                                                       float* __restrict__ mean,
                                                       float* __restrict__ scale) {
    const int c = threadIdx.x;   // 128 threads
    float s = 0.f, q = 0.f;
    for (int b = 0; b < NB_STAT; ++b) {
        s += psum[(size_t)b * D_H + c];
        q += psq [(size_t)b * D_H + c];
    }
    const float m = s * (1.0f / (float)N_NODES);
    const float v = q * (1.0f / (float)N_NODES) - m * m;
    mean[c]  = m;
    scale[c] = rsqrtf(v + BN_EPS);
}

// ---------------------------------------------------------------------------
extern "C" void kernel_launch(void* const* d_in, const int* in_sizes, int n_in,
                              void* d_out, int out_size, void* d_ws, size_t ws_size,
                              hipStream_t stream) {
    const float* x        = (const float*)d_in[0];
    const float* W0       = (const float*)d_in[1];
    const float* b0       = (const float*)d_in[2];
    const float* W1       = (const float*)d_in[3];
    const float* b1       = (const float*)d_in[4];
    const float* adj_vals = (const float*)d_in[5];
    const int*   row      = (const int*)d_in[6];
    const int*   col      = (const int*)d_in[7];
    float*       out      = (float*)d_out;

    // Workspace layout (floats): h0 | h1 | g1 | psum | psq | mean | scale
    float* ws    = (float*)d_ws;
    float* h0    = ws;                                      // 100000*128
    float* h1    = h0   + (size_t)N_NODES * D_H;            // 100000*128
    float* g1    = h1   + (size_t)N_NODES * D_H;            // 100000*64
    float* psum  = g1   + (size_t)N_NODES * D_OUT;          // NB_STAT*128
    float* psq   = psum + (size_t)NB_STAT * D_H;            // NB_STAT*128
    float* meanv = psq  + (size_t)NB_STAT * D_H;            // 128
    float* scalv = meanv + D_H;                             // 128

    const int g0_blocks = (MTILES + 7) / 8;                 // 782 (8 waves/block)
    const int g1_blocks = (MTILES * 32 + 255) / 256;        // 782

    gcn_gemm0<<<g0_blocks, 256, 0, stream>>>(x, W0, h0);
    gcn_spmm<D_H, true><<<N_NODES, D_H, 0, stream>>>(adj_vals, row, col, h0, b0, h1);
    gcn_bn_stats<<<NB_STAT, 256, 0, stream>>>(h1, psum, psq);
    gcn_bn_finalize<<<1, D_H, 0, stream>>>(psum, psq, meanv, scalv);
    gcn_gemm1<<<g1_blocks, 256, 0, stream>>>(h1, meanv, scalv, W1, g1);
    gcn_spmm<D_OUT, false><<<N_NODES, D_OUT, 0, stream>>>(adj_vals, row, col, g1, b1, out);
}